// Rotation45SymmetricPosEmbed_77781857731047
// MI455X (gfx1250) — compile-verified
//
#include <hip/hip_runtime.h>
#include <math.h>

// Problem constants (match the reference: H=32, C=128, B=128, L=136 wedge positions)
#define HGRID   32
#define CCH     128
#define NBATCH  128
#define NROW    1025        // 1 cls row + 32*32 patch rows
#define ROWF    1024        // 8*C floats per row
#define LPOS    136         // positions in the 0..pi/4 wedge of a 32x32 grid
#define NCOMBO  (LPOS * 8)  // (idx, rotation) pairs
#define YSPLIT  8           // batch split across gridDim.y
#define BPER    (NBATCH / YSPLIT)

// ---- CDNA5-specific data movement helpers (gfx1250 ISA) -------------------

// GLOBAL_LOAD_ASYNC_TO_LDS_B128: per-lane 16B global -> LDS, tracked by ASYNCcnt.
// VDST = LDS byte address VGPR, VADDR = 64-bit global address VGPR pair.
__device__ __forceinline__ void async_ld_b128(const void* gptr, unsigned lds_off) {
  asm volatile("global_load_async_to_lds_b128 %0, %1, off"
               :: "v"(lds_off), "v"(gptr)
               : "memory");
}

__device__ __forceinline__ void wait_async0() {
  asm volatile("s_wait_asynccnt 0" ::: "memory");
}

// ---------------------------------------------------------------------------

__global__ __launch_bounds__(256)
void posemb_add_kernel(const float* __restrict__ x,     // (128, 1025, 1024)
                       const float* __restrict__ P,     // (136, 8, 128) flat
                       const float* __restrict__ cls,   // (128,)
                       float* __restrict__ out) {       // (128, 1025, 1024)
  __shared__ __align__(16) float s_row[ROWF];  // staged embedding row (4 KB)
  __shared__ int s_key;

  const int t = threadIdx.x;     // 0..255; owns float4 at row offset 4*t
  const int n = blockIdx.x;      // row of `full`: 0 = cls, 1..1024 = grid cell

  if (t == 0) s_key = -1;
  __syncthreads();

  float4 e;
  e.x = 0.f; e.y = 0.f; e.z = 0.f; e.w = 0.f;

  if (n == 0) {
    // cls row: full[0, k*128+c] = cls[c]  (tile x8)
    e = *(const float4*)(cls + ((4 * t) & (CCH - 1)));
  } else {
    const int ci = (n - 1) >> 5;   // grid i
    const int cj = (n - 1) & 31;   // grid j

    // Derive (_SRC, _ROT, _MASK)[ci][cj] by scanning all 1088 forward
    // rotation mappings; "last write wins" == max (idx*8 + k).
    // Double math mirrors the Python reference exactly (no FP contraction,
    // rint == round-half-even == Python round, literal math.pi, (k*pi)/4).
    int lk = -1;
    for (int q = t; q < NCOMBO; q += 256) {
      const int idx = q >> 3;
      const int k   = q & 7;

      // idx -> wedge position (pi, pj): row i holds i+1 cells, j = 31-i+r
      int pi = 0, t0 = 0;
      while (t0 + pi + 1 <= idx) { t0 += pi + 1; ++pi; }
      const int pj = 31 - pi + (idx - t0);

      const double yy = __dsub_rn(15.5, (double)pi);
      const double xx = __dsub_rn((double)pj, 15.5);
      const double th = ((double)k * 3.141592653589793) / 4.0;
      const double ct = cos(th);
      const double st = sin(th);
      const double xn = __dsub_rn(__dmul_rn(ct, xx), __dmul_rn(st, yy));
      const double yn = __dadd_rn(__dmul_rn(st, xx), __dmul_rn(ct, yy));

      int ir = (int)rint(__dsub_rn(15.5, yn));
      int jr = (int)rint(__dadd_rn(15.5, xn));
      ir = ir < 0 ? 0 : (ir > 31 ? 31 : ir);
      jr = jr < 0 ? 0 : (jr > 31 ? 31 : jr);

      if (ir == ci && jr == cj) lk = q;   // q ascends -> keeps the last match
    }
    if (lk >= 0) atomicMax(&s_key, lk);
    __syncthreads();

    const int key = s_key;
    if (key >= 0) {                       // block-uniform branch (mask)
      const int src = key >> 3;
      const int rot = key & 7;
      const int kk  = t >> 5;                         // eighth index of this float4
      const int g   = (kk - rot + 8) & 7;             // (k - rot) mod 8
      const float* pp = P + ((size_t)src * ROWF + (size_t)g * CCH + ((4 * t) & (CCH - 1)));

      // CDNA5 async tensor path: gather 16B/lane into LDS, wait ASYNCcnt.
      const unsigned lds_off = (unsigned)(size_t)(&s_row[4 * t]);
      async_ld_b128(pp, lds_off);
      wait_async0();
      e = *(const float4*)(&s_row[4 * t]);
    }
    // else: masked cell -> embedding row is zeros (e already zero)
  }

  // Bandwidth-bound phase: out[b, n, :] = x[b, n, :] + e, for 16 images.
  const size_t img = (size_t)NROW * ROWF;               // image stride in floats
  const size_t base = ((size_t)(blockIdx.y * BPER) * NROW + (size_t)n) * ROWF + (size_t)(4 * t);
  const float* xp = x + base;
  float*       op = out + base;

#pragma unroll 4
  for (int b = 0; b < BPER; ++b) {
    __builtin_prefetch(xp + img, 0, 3);   // global_prefetch_b8: prime next image
    float4 v = *(const float4*)xp;        // global_load_b128
    v.x += e.x; v.y += e.y; v.z += e.z; v.w += e.w;
    *(float4*)op = v;                     // global_store_b128
    xp += img;
    op += img;
  }
}

extern "C" void kernel_launch(void* const* d_in, const int* in_sizes, int n_in,
                              void* d_out, int out_size, void* d_ws, size_t ws_size,
                              hipStream_t stream) {
  (void)in_sizes; (void)n_in; (void)out_size; (void)d_ws; (void)ws_size;
  const float* x   = (const float*)d_in[0];   // (128, 1025, 1024) fp32
  const float* P   = (const float*)d_in[1];   // (1, 136, 1024) fp32
  const float* cls = (const float*)d_in[2];   // (1, 1, 128) fp32
  float* out = (float*)d_out;

  dim3 grid(NROW, YSPLIT);
  posemb_add_kernel<<<grid, 256, 0, stream>>>(x, P, cls, out);
}